// DFT_10316511445664
// MI455X (gfx1250) — compile-verified
//
#include <hip/hip_runtime.h>

typedef float v2f __attribute__((ext_vector_type(2)));
typedef float v8f __attribute__((ext_vector_type(8)));

#define NFFT        100
#define CROP        50
#define NCOL        112          // 100 cols padded to 7 tiles of 16
#define NTILES      7
#define KSTEPS      25           // 100 / 4
#define ROWS_PER_WG 128
#define THREADS     256
#define NEG_HALF_D_LOG2PI (-91.893853320467274f)  // -0.5 * 100 * ln(2*pi)

__global__ __launch_bounds__(THREADS)
void dft_mvn_kernel(const float* __restrict__ x,
                    float* __restrict__ out,      // [B, 100] = [B, 2, 50] x_fft
                    float* __restrict__ logpz,    // [B]
                    int Brows)
{
    // W stored K-innermost: sW[j*100 + n] = W_T[n][j]  (44.8 KB)
    __shared__ float sW[NCOL * NFFT];
    // x rows, stride 100 floats (bank-conflict free for 16-row A frags) (51.2 KB)
    __shared__ float sX[ROWS_PER_WG * NFFT];

    const int tid  = threadIdx.x;
    const int lane = tid & 31;
    const int lo   = lane & 15;    // column / row-in-tile selector
    const int hi   = lane >> 4;    // K/M half selector
    const int wave = tid >> 5;
    const int row0 = blockIdx.x * ROWS_PER_WG;
    const int rowsValid = Brows - row0;

    // ---- Build the DFT twiddle matrix in LDS ----
    const float inv100 = 0.01f;
    const float twopi  = 6.28318530717958647692f;
    for (int t = tid; t < NCOL * NFFT; t += THREADS) {
        const int j = t / NFFT;       // output column 0..111
        const int n = t - j * NFFT;   // sample index 0..99
        float v = 0.0f;
        if (j < NFFT) {
            const int freq = (j < CROP) ? (j + CROP) : j;
            const int ph   = (freq * n) % NFFT;            // exact phase mod 100
            const float ang = twopi * (float)ph * inv100;
            v = ((j < CROP) ? __cosf(ang) : -__sinf(ang)) * inv100;
        }
        sW[t] = v;
    }

    // ---- Stage 128 rows of x into LDS (coalesced) ----
    {
        const float* xrow = x + (size_t)row0 * NFFT;
        for (int t = tid; t < ROWS_PER_WG * NFFT; t += THREADS) {
            const int r = t / NFFT;
            sX[t] = (r < rowsValid) ? xrow[t] : 0.0f;
        }
    }
    __syncthreads();

    // ---- Per-wave 16-row strip, 7 column tiles, K=100 in 25 steps of 4 ----
    const int mybase = wave * 16;     // local row base
    v8f acc[NTILES] = {};

    for (int ks = 0; ks < KSTEPS; ++ks) {
        const int kb = ks * 4 + hi * 2;   // ISA A/B f32 layout: half-wave holds K pair {0,1}/{2,3}
        // A fragment: A[M = mybase+lo][K = kb, kb+1]  (8B aligned -> ds_load_b64)
        const v2f a = *(const v2f*)&sX[(mybase + lo) * NFFT + kb];
        #pragma unroll
        for (int t = 0; t < NTILES; ++t) {
            // B fragment: B[K = kb, kb+1][N = 16t+lo]
            const v2f b = *(const v2f*)&sW[(t * 16 + lo) * NFFT + kb];
            acc[t] = __builtin_amdgcn_wmma_f32_16x16x4_f32(
                false, a, false, b, (short)0, acc[t], false, false);
        }
    }

    // ---- Store x_fft: D layout lane holds D[M = r + 8*hi][N = lo] in elem r ----
    const size_t gr = (size_t)row0 + (size_t)mybase;
    #pragma unroll
    for (int t = 0; t < NTILES; ++t) {
        const int j = t * 16 + lo;
        if (j < NFFT) {
            #pragma unroll
            for (int r = 0; r < 8; ++r) {
                const size_t m = gr + (size_t)(r + hi * 8);
                if (m < (size_t)Brows)
                    out[m * NFFT + j] = acc[t][r];
            }
        }
    }

    // ---- Fused MVN log-prob: sum of squares over the 100 columns per row ----
    #pragma unroll
    for (int r = 0; r < 8; ++r) {
        float s = 0.0f;
        #pragma unroll
        for (int t = 0; t < NTILES; ++t) {
            const int j = t * 16 + lo;
            const float v = (j < NFFT) ? acc[t][r] : 0.0f;
            s += v * v;
        }
        // reduce across the 16 lanes of this half-wave (wave32 shuffles)
        s += __shfl_xor(s, 1, 32);
        s += __shfl_xor(s, 2, 32);
        s += __shfl_xor(s, 4, 32);
        s += __shfl_xor(s, 8, 32);
        if (lo == 0) {
            const size_t m = gr + (size_t)(r + hi * 8);
            if (m < (size_t)Brows)
                logpz[m] = -0.5f * s + NEG_HALF_D_LOG2PI;
        }
    }
}

extern "C" void kernel_launch(void* const* d_in, const int* in_sizes, int n_in,
                              void* d_out, int out_size, void* d_ws, size_t ws_size,
                              hipStream_t stream) {
    const float* x = (const float*)d_in[0];
    const int B = in_sizes[0] / NFFT;          // 262144
    float* out   = (float*)d_out;              // first B*100 floats: x_fft
    float* logpz = out + (size_t)B * NFFT;     // next B floats: log_pz
    const int blocks = (B + ROWS_PER_WG - 1) / ROWS_PER_WG;
    hipLaunchKernelGGL(dft_mvn_kernel, dim3(blocks), dim3(THREADS), 0, stream,
                       x, out, logpz, B);
}